// SGA_75531294867605
// MI455X (gfx1250) — compile-verified
//
#include <hip/hip_runtime.h>
#include <hip/hip_bf16.h>
#include <math.h>

#define DEVINL __device__ __forceinline__

typedef __attribute__((ext_vector_type(2))) float v2f;
typedef __attribute__((ext_vector_type(8))) float v8f;

// ---------------- WMMA helpers (fp32 16x16x4) ----------------
DEVINL v8f wmma4(v2f a, v2f b, v8f c) {
  return __builtin_amdgcn_wmma_f32_16x16x4_f32(false, a, false, b, (short)0, c,
                                               false, false);
}

DEVINL v8f vzero8() { v8f z = {0.f,0.f,0.f,0.f,0.f,0.f,0.f,0.f}; return z; }

// A/B fragment from a row-major matrix: element ((row0 + l%16), (k0 + 2*(l>>4)+{0,1}))
// Works for A (16xK tile rows) and for B of an NT GEMM (B[k][n] = M[n_row][k]).
DEVINL v2f frag_ld_g(const float* __restrict__ p, int row0, int k0, int ld) {
  const int l = threadIdx.x & 31;
  const float* q = p + (size_t)(row0 + (l & 15)) * ld + k0 + ((l >> 4) << 1);
  v2f r; r.x = q[0]; r.y = q[1];
  return r;
}

// Same pattern but from LDS (16-row tile, row stride ld).
DEVINL v2f frag_ld_s(const float* p, int k0, int ld) {
  const int l = threadIdx.x & 31;
  const float* q = p + (l & 15) * ld + k0 + ((l >> 4) << 1);
  v2f r; r.x = q[0]; r.y = q[1];
  return r;
}

DEVINL float sigm(float x) { return 1.f / (1.f + __expf(-x)); }
DEVINL float lrelu(float x) { return x > 0.f ? x : 0.2f * x; }

// ---------------- Fused 2-layer GRU (per-wave 16-row tile, states in LDS) ----------------
// grid = 256 blocks of 32 threads (1 wave); each wave advances 16 batch rows
// through both GRU layers for all 64 timesteps, writes only the final h1 -> g.
__global__ void __launch_bounds__(32) gru_kernel(
    const float* __restrict__ x,
    const float* __restrict__ Wih0, const float* __restrict__ Whh0,
    const float* __restrict__ bih0, const float* __restrict__ bhh0,
    const float* __restrict__ Wih1, const float* __restrict__ Whh1,
    const float* __restrict__ bih1, const float* __restrict__ bhh1,
    float* __restrict__ g) {
  __shared__ float xb[16 * 32];
  __shared__ float h0[2][16 * 128];
  __shared__ float h1[2][16 * 128];
  const int l = threadIdx.x;
  const int row0 = blockIdx.x * 16;
  const int nloc = l & 15;
  const int m0 = (l >> 4) << 3;

  for (int i = l; i < 16 * 128; i += 32) { h0[0][i] = 0.f; h1[0][i] = 0.f; }
  int cur = 0;

  for (int t = 0; t < 64; ++t) {
    const int nxt = cur ^ 1;
    // stage x_t tile (16 x 32) into LDS
    for (int i = l; i < 16 * 32; i += 32) {
      const int m = i >> 5, k = i & 31;
      xb[i] = x[((size_t)(row0 + m) * 64 + t) * 32 + k];
    }
    // ---------------- layer 0 ----------------
    for (int c = 0; c < 8; ++c) {
      v8f ar = vzero8(), az = vzero8(), ain = vzero8(), ahn = vzero8();
      for (int k = 0; k < 32; k += 4) {               // x part (K = F = 32)
        v2f a  = frag_ld_s(xb, k, 32);
        v2f br = frag_ld_g(Wih0,        c * 16, k, 32);
        v2f bz = frag_ld_g(Wih0, 128 +  c * 16, k, 32);
        v2f bn = frag_ld_g(Wih0, 256 +  c * 16, k, 32);
        ar = wmma4(a, br, ar); az = wmma4(a, bz, az); ain = wmma4(a, bn, ain);
      }
      for (int k = 0; k < 128; k += 4) {              // h part (K = H = 128)
        v2f a  = frag_ld_s(h0[cur], k, 128);
        v2f br = frag_ld_g(Whh0,        c * 16, k, 128);
        v2f bz = frag_ld_g(Whh0, 128 +  c * 16, k, 128);
        v2f bn = frag_ld_g(Whh0, 256 +  c * 16, k, 128);
        ar = wmma4(a, br, ar); az = wmma4(a, bz, az); ahn = wmma4(a, bn, ahn);
      }
      const int j = c * 16 + nloc;
      const float br_  = bih0[j]       + bhh0[j];
      const float bz_  = bih0[128 + j] + bhh0[128 + j];
      const float bin_ = bih0[256 + j];
      const float bhn_ = bhh0[256 + j];
#pragma unroll
      for (int r = 0; r < 8; ++r) {
        const int m = m0 + r;
        const float hp = h0[cur][m * 128 + j];
        const float rg = sigm(ar[r] + br_);
        const float zg = sigm(az[r] + bz_);
        const float ng = tanhf(ain[r] + bin_ + rg * (ahn[r] + bhn_));
        h0[nxt][m * 128 + j] = (1.f - zg) * ng + zg * hp;
      }
    }
    // ---------------- layer 1 (x = h0[nxt]) ----------------
    for (int c = 0; c < 8; ++c) {
      v8f ar = vzero8(), az = vzero8(), ain = vzero8(), ahn = vzero8();
      for (int k = 0; k < 128; k += 4) {
        v2f a  = frag_ld_s(h0[nxt], k, 128);
        v2f br = frag_ld_g(Wih1,        c * 16, k, 128);
        v2f bz = frag_ld_g(Wih1, 128 +  c * 16, k, 128);
        v2f bn = frag_ld_g(Wih1, 256 +  c * 16, k, 128);
        ar = wmma4(a, br, ar); az = wmma4(a, bz, az); ain = wmma4(a, bn, ain);
      }
      for (int k = 0; k < 128; k += 4) {
        v2f a  = frag_ld_s(h1[cur], k, 128);
        v2f br = frag_ld_g(Whh1,        c * 16, k, 128);
        v2f bz = frag_ld_g(Whh1, 128 +  c * 16, k, 128);
        v2f bn = frag_ld_g(Whh1, 256 +  c * 16, k, 128);
        ar = wmma4(a, br, ar); az = wmma4(a, bz, az); ahn = wmma4(a, bn, ahn);
      }
      const int j = c * 16 + nloc;
      const float br_  = bih1[j]       + bhh1[j];
      const float bz_  = bih1[128 + j] + bhh1[128 + j];
      const float bin_ = bih1[256 + j];
      const float bhn_ = bhh1[256 + j];
#pragma unroll
      for (int r = 0; r < 8; ++r) {
        const int m = m0 + r;
        const float hp = h1[cur][m * 128 + j];
        const float rg = sigm(ar[r] + br_);
        const float zg = sigm(az[r] + bz_);
        const float ng = tanhf(ain[r] + bin_ + rg * (ahn[r] + bhn_));
        h1[nxt][m * 128 + j] = (1.f - zg) * ng + zg * hp;
      }
    }
    cur = nxt;
  }
  for (int i = l; i < 16 * 128; i += 32)
    g[(size_t)(row0 + (i >> 7)) * 128 + (i & 127)] = h1[cur][i];
}

// ---------------- row L2-normalize (+ diag validity flag) ----------------
__global__ void __launch_bounds__(128) rownorm_kernel(
    const float* __restrict__ in, float* __restrict__ outhat, float* __restrict__ dv) {
  __shared__ float s[128];
  const int i = blockIdx.x, t = threadIdx.x;
  const float v = in[(size_t)i * 128 + t];
  s[t] = v * v; __syncthreads();
  for (int st = 64; st > 0; st >>= 1) { if (t < st) s[t] += s[t + st]; __syncthreads(); }
  const float nrm = sqrtf(s[0]);
  const float inv = (nrm > 0.f) ? 1.f / nrm : 0.f;
  outhat[(size_t)i * 128 + t] = v * inv;
  if (dv != nullptr && t == 0) dv[i] = (nrm > 0.f) ? 1.f : 0.f;
}

// ---------------- C = A * B^T, A:(M x 128), B:(rows x 128) row-major, WMMA ----------------
// grid.x = M/16 row tiles, grid.y = col groups of 128; 1 wave per block.
__global__ void __launch_bounds__(32) gemm_nt_kernel(
    const float* __restrict__ A, const float* __restrict__ B,
    float* __restrict__ C, int ldc) {
  const int i0 = blockIdx.x * 16;
  const int j0 = blockIdx.y * 128;
  v8f acc[8];
#pragma unroll
  for (int tb = 0; tb < 8; ++tb) acc[tb] = vzero8();
  for (int k = 0; k < 128; k += 4) {
    v2f a = frag_ld_g(A, i0, k, 128);
#pragma unroll
    for (int tb = 0; tb < 8; ++tb) {
      v2f b = frag_ld_g(B, j0 + tb * 16, k, 128);
      acc[tb] = wmma4(a, b, acc[tb]);
    }
  }
  const int l = threadIdx.x;
  const int n0 = l & 15, m0 = (l >> 4) << 3;
#pragma unroll
  for (int tb = 0; tb < 8; ++tb)
#pragma unroll
    for (int r = 0; r < 8; ++r)
      C[(size_t)(i0 + m0 + r) * ldc + j0 + tb * 16 + n0] = acc[tb][r];
}

// ---------------- per-row top-K (K=20) by |sim|, diag excluded ----------------
__global__ void __launch_bounds__(256) topk_kernel(
    const float* __restrict__ sim, int* __restrict__ cols, float* __restrict__ vals) {
  __shared__ float sv[4096];
  __shared__ float rv[256];
  __shared__ int   ri[256];
  const int i = blockIdx.x, t = threadIdx.x;
  const float* row = sim + (size_t)i * 4096;
  for (int j = t; j < 4096; j += 256) sv[j] = (j == i) ? -1.f : fabsf(row[j]);
  __syncthreads();
  for (int k = 0; k < 20; ++k) {
    float bv = -3.f; int bi = 0;
    for (int j = t; j < 4096; j += 256) {
      const float v = sv[j];
      if (v > bv) { bv = v; bi = j; }
    }
    rv[t] = bv; ri[t] = bi; __syncthreads();
    for (int st = 128; st > 0; st >>= 1) {
      if (t < st) {
        if (rv[t + st] > rv[t] || (rv[t + st] == rv[t] && ri[t + st] < ri[t])) {
          rv[t] = rv[t + st]; ri[t] = ri[t + st];
        }
      }
      __syncthreads();
    }
    if (t == 0) {
      const int c = ri[0];
      cols[i * 20 + k] = c;
      vals[i * 20 + k] = row[c];
      sv[c] = -2.f;
    }
    __syncthreads();
  }
}

__global__ void zero_kernel(float* __restrict__ p, int n) {
  const int i = blockIdx.x * blockDim.x + threadIdx.x;
  if (i < n) p[i] = 0.f;
}

// ---------------- sparse tks^T @ g via atomics (K=20 per row) ----------------
__global__ void __launch_bounds__(128) scatter_kernel(
    const int* __restrict__ cols, const float* __restrict__ vals,
    const float* __restrict__ g, float* __restrict__ cpt, float* __restrict__ colsum) {
  const int i = blockIdx.x, t = threadIdx.x;
  const float gv = g[(size_t)i * 128 + t];
  for (int k = 0; k < 20; ++k) {
    const int c = cols[i * 20 + k];
    const float v = vals[i * 20 + k];
    atomicAdd(&cpt[(size_t)c * 128 + t], v * gv);
    if (t == 0) atomicAdd(&colsum[c], v);
  }
}

// add diag term where column was touched; compute validity of each concept row
__global__ void __launch_bounds__(128) cptfin_kernel(
    float* __restrict__ cpt, const float* __restrict__ colsum,
    const float* __restrict__ dv, const float* __restrict__ g,
    float* __restrict__ valid) {
  __shared__ float s[128];
  const int j = blockIdx.x, t = threadIdx.x;
  float v = cpt[(size_t)j * 128 + t];
  if (colsum[j] != 0.f) v += dv[j] * g[(size_t)j * 128 + t];
  cpt[(size_t)j * 128 + t] = v;
  s[t] = v; __syncthreads();
  for (int st = 64; st > 0; st >>= 1) { if (t < st) s[t] += s[t + st]; __syncthreads(); }
  if (t == 0) valid[j] = (s[0] != 0.f) ? 1.f : 0.f;
}

// cf = lrelu(cpt @ W_hc^T + b_hc) * valid[row]   (WMMA NT GEMM + epilogue)
__global__ void __launch_bounds__(32) cf_kernel(
    const float* __restrict__ cpt, const float* __restrict__ Whc,
    const float* __restrict__ bhc, const float* __restrict__ valid,
    float* __restrict__ cf) {
  const int i0 = blockIdx.x * 16;
  v8f acc[8];
#pragma unroll
  for (int tb = 0; tb < 8; ++tb) acc[tb] = vzero8();
  for (int k = 0; k < 128; k += 4) {
    v2f a = frag_ld_g(cpt, i0, k, 128);
#pragma unroll
    for (int tb = 0; tb < 8; ++tb) {
      v2f b = frag_ld_g(Whc, tb * 16, k, 128);
      acc[tb] = wmma4(a, b, acc[tb]);
    }
  }
  const int l = threadIdx.x;
  const int n0 = l & 15, m0 = (l >> 4) << 3;
#pragma unroll
  for (int tb = 0; tb < 8; ++tb) {
    const float bb = bhc[tb * 16 + n0];
#pragma unroll
    for (int r = 0; r < 8; ++r) {
      const int row = i0 + m0 + r;
      cf[(size_t)row * 128 + tb * 16 + n0] = lrelu(acc[tb][r] + bb) * valid[row];
    }
  }
}

// ---------------- column softmax stats (online logsumexp over rows) ----------------
__global__ void __launch_bounds__(256) colstats_kernel(
    const float* __restrict__ s2, float* __restrict__ mcol, float* __restrict__ dcol) {
  __shared__ float ms[256], ds[256];
  const int j0 = blockIdx.x * 16;
  const int c = threadIdx.x & 15, rs = threadIdx.x >> 4;
  float m = -1e30f, d = 0.f;
  for (int i = rs; i < 4096; i += 16) {
    const float v = s2[(size_t)i * 4096 + j0 + c];
    const float mn = fmaxf(m, v);
    d = d * __expf(m - mn) + __expf(v - mn);
    m = mn;
  }
  ms[threadIdx.x] = m; ds[threadIdx.x] = d; __syncthreads();
  for (int st = 8; st > 0; st >>= 1) {
    if (rs < st) {
      const float m2 = ms[(rs + st) * 16 + c], d2 = ds[(rs + st) * 16 + c];
      const float mn = fmaxf(m, m2);
      d = d * __expf(m - mn) + d2 * __expf(m2 - mn);
      m = mn;
      ms[threadIdx.x] = m; ds[threadIdx.x] = d;
    }
    __syncthreads();
  }
  if (rs == 0) { mcol[j0 + c] = m; dcol[j0 + c] = d; }
}

// ---------------- hs_acc += softmax(s2,axis=0) @ cf2, K-split, on-the-fly A ----------------
__global__ void __launch_bounds__(32) attn_gemm_kernel(
    const float* __restrict__ s2, const float* __restrict__ mcol,
    const float* __restrict__ dcol, const float* __restrict__ cf2,
    float* __restrict__ hs) {
  const int i0 = blockIdx.x * 16;
  const int kbeg = blockIdx.y * 512, kend = kbeg + 512;
  const int l = threadIdx.x;
  const int mi = l & 15, kh = (l >> 4) << 1;
  v8f acc[8];
#pragma unroll
  for (int tb = 0; tb < 8; ++tb) acc[tb] = vzero8();
  for (int k = kbeg; k < kend; k += 4) {
    const int j = k + kh;                      // this lane's two K (=column j) indices
    const float* sp = s2 + (size_t)(i0 + mi) * 4096 + j;
    v2f a;
    a.x = __expf(sp[0] - mcol[j])     / dcol[j];
    a.y = __expf(sp[1] - mcol[j + 1]) / dcol[j + 1];
#pragma unroll
    for (int tb = 0; tb < 8; ++tb) {
      const int n = tb * 16 + mi;
      v2f b;
      b.x = cf2[(size_t)j * 128 + n];
      b.y = cf2[(size_t)(j + 1) * 128 + n];
      acc[tb] = wmma4(a, b, acc[tb]);
    }
  }
  const int m0 = (l >> 4) << 3;
#pragma unroll
  for (int tb = 0; tb < 8; ++tb)
#pragma unroll
    for (int r = 0; r < 8; ++r)
      atomicAdd(&hs[(size_t)(i0 + m0 + r) * 128 + tb * 16 + mi], acc[tb][r]);
}

__global__ void __launch_bounds__(128) hsepi_kernel(
    float* __restrict__ hs, const float* __restrict__ bhs) {
  const int i = blockIdx.x, t = threadIdx.x;
  hs[(size_t)i * 128 + t] = lrelu(hs[(size_t)i * 128 + t] + bhs[t]);
}

// ---------------- fused output heads: back/fore/indi/pred ----------------
__global__ void __launch_bounds__(128) heads_kernel(
    const float* __restrict__ hs, const float* __restrict__ g,
    const float* __restrict__ Wf, const float* __restrict__ bf,
    const float* __restrict__ Wb, const float* __restrict__ bb,
    const float* __restrict__ Wi, const float* __restrict__ bi,
    const float* __restrict__ Wo, const float* __restrict__ bo,
    float* __restrict__ pred) {
  __shared__ float hrow[128], gmb[128], rs[128];
  const int i = blockIdx.x, t = threadIdx.x;
  hrow[t] = hs[(size_t)i * 128 + t]; __syncthreads();
  float ab = 0.f, af = 0.f;
  for (int k = 0; k < 128; ++k) {
    const float hv = hrow[k];
    ab += hv * Wb[t * 128 + k];
    af += hv * Wf[t * 128 + k];
  }
  const float back = lrelu(ab + bb[t]);
  const float fore = lrelu(af + bf[t]);
  gmb[t] = g[(size_t)i * 128 + t] - back; __syncthreads();
  float ai = 0.f;
  for (int k = 0; k < 128; ++k) ai += gmb[k] * Wi[t * 128 + k];
  const float indi = lrelu(ai + bi[t]);
  rs[t] = (fore + indi) * Wo[t]; __syncthreads();
  for (int st = 64; st > 0; st >>= 1) { if (t < st) rs[t] += rs[t + st]; __syncthreads(); }
  if (t == 0) pred[i] = rs[0] + bo[0];
}

// ---------------- launcher ----------------
extern "C" void kernel_launch(void* const* d_in, const int* in_sizes, int n_in,
                              void* d_out, int out_size, void* d_ws, size_t ws_size,
                              hipStream_t stream) {
  const float* x      = (const float*)d_in[0];
  const float* Wih0   = (const float*)d_in[1];
  const float* Whh0   = (const float*)d_in[2];
  const float* bih0   = (const float*)d_in[3];
  const float* bhh0   = (const float*)d_in[4];
  const float* Wih1   = (const float*)d_in[5];
  const float* Whh1   = (const float*)d_in[6];
  const float* bih1   = (const float*)d_in[7];
  const float* bhh1   = (const float*)d_in[8];
  const float* W_hc   = (const float*)d_in[9];
  const float* b_hc   = (const float*)d_in[10];
  const float* W_hs   = (const float*)d_in[11];
  const float* b_hs   = (const float*)d_in[12];
  const float* W_fore = (const float*)d_in[13];
  const float* b_fore = (const float*)d_in[14];
  const float* W_back = (const float*)d_in[15];
  const float* b_back = (const float*)d_in[16];
  const float* W_indi = (const float*)d_in[17];
  const float* b_indi = (const float*)d_in[18];
  const float* W_out  = (const float*)d_in[19];
  const float* b_out  = (const float*)d_in[20];
  float* out = (float*)d_out;

  const size_t N = 4096, H = 128, NH = N * H;
  float* ws = (float*)d_ws;
  size_t o = 0;
  float* g_     = ws + o; o += NH;
  float* ghat   = ws + o; o += NH;
  float* dv     = ws + o; o += N;
  float* simb   = ws + o; o += N * N;      // reused for s2
  int*   colsI  = (int*)(ws + o); o += N * 20;
  float* vals   = ws + o; o += N * 20;
  float* colsum = ws + o; o += N;
  float* cpt    = ws + o; o += NH;
  float* validv = ws + o; o += N;
  float* cfv    = ws + o; o += NH;
  float* cfhat  = ws + o; o += NH;
  float* mcol   = ws + o; o += N;
  float* dcol   = ws + o; o += N;
  float* cf2    = ws + o; o += NH;
  float* hsv    = ws + o; o += NH;

  // 1) fused 2-layer GRU -> g (N x H)
  gru_kernel<<<256, 32, 0, stream>>>(x, Wih0, Whh0, bih0, bhh0,
                                     Wih1, Whh1, bih1, bhh1, g_);
  // 2) normalize g; diag validity
  rownorm_kernel<<<4096, 128, 0, stream>>>(g_, ghat, dv);
  // 3) sim = ghat @ ghat^T (N x N)
  gemm_nt_kernel<<<dim3(256, 32), 32, 0, stream>>>(ghat, ghat, simb, 4096);
  // 4) per-row top-20 by |sim| (diag excluded)
  topk_kernel<<<4096, 256, 0, stream>>>(simb, colsI, vals);
  // 5) zero accumulators
  zero_kernel<<<(int)((NH + 255) / 256), 256, 0, stream>>>(cpt, (int)NH);
  zero_kernel<<<(int)((N + 255) / 256), 256, 0, stream>>>(colsum, (int)N);
  zero_kernel<<<(int)((NH + 255) / 256), 256, 0, stream>>>(hsv, (int)NH);
  // 6) concept_raw = tks^T @ g  (sparse scatter)
  scatter_kernel<<<4096, 128, 0, stream>>>(colsI, vals, g_, cpt, colsum);
  // 7) add diag(col_nz * diag) @ g; compute valid
  cptfin_kernel<<<4096, 128, 0, stream>>>(cpt, colsum, dv, g_, validv);
  // 8) cf = lrelu(concept @ W_hc^T + b_hc) * valid
  cf_kernel<<<256, 32, 0, stream>>>(cpt, W_hc, b_hc, validv, cfv);
  // 9) normalize cf
  rownorm_kernel<<<4096, 128, 0, stream>>>(cfv, cfhat, nullptr);
  // 10) s2 = ghat @ cfhat^T (reuse sim buffer)
  gemm_nt_kernel<<<dim3(256, 32), 32, 0, stream>>>(ghat, cfhat, simb, 4096);
  // 11) column-wise softmax stats
  colstats_kernel<<<256, 256, 0, stream>>>(simb, mcol, dcol);
  // 12) cf2 = cf @ W_hs^T
  gemm_nt_kernel<<<dim3(256, 1), 32, 0, stream>>>(cfv, W_hs, cf2, 128);
  // 13) hs_acc = softmax(s2, axis=0) @ cf2 (K-split, atomic accumulate)
  attn_gemm_kernel<<<dim3(256, 8), 32, 0, stream>>>(simb, mcol, dcol, cf2, hsv);
  // 14) hs = lrelu(hs_acc + b_hs)
  hsepi_kernel<<<4096, 128, 0, stream>>>(hsv, b_hs);
  // 15) back/fore/indi heads + final projection
  heads_kernel<<<4096, 128, 0, stream>>>(hsv, g_, W_fore, b_fore, W_back, b_back,
                                         W_indi, b_indi, W_out, b_out, out);
  (void)in_sizes; (void)n_in; (void)out_size; (void)ws_size;
}